// DDSL_spec_82772609728632
// MI455X (gfx1250) — compile-verified
//
#include <hip/hip_runtime.h>
#include <math.h>

typedef __attribute__((ext_vector_type(2))) float v2f;
typedef __attribute__((ext_vector_type(8))) float v8f;

#define N_ELEM   256
#define N_VERT   160
#define RES0     256
#define RES1     129                 // rfft bins on last axis
#define NBINS    (RES0 * RES1)       // 33024
#define TWO_PI_F 6.283185307179586f

// ------------------------------------------------------------------
// Kernel 1: gather P = V[E] (padded to 4 rows/element) and CD = C*D
// ------------------------------------------------------------------
__global__ __launch_bounds__(256) void ddsl_precompute(
    const float* __restrict__ V, const int* __restrict__ E,
    const float* __restrict__ D,
    float* __restrict__ Px, float* __restrict__ Py, float* __restrict__ CD)
{
    int e = blockIdx.x * blockDim.x + threadIdx.x;
    if (e >= N_ELEM) return;
    int i0 = E[e * 3 + 0];
    int i1 = E[e * 3 + 1];
    int i2 = E[e * 3 + 2];
    float x0 = V[2 * i0], y0 = V[2 * i0 + 1];
    float x1 = V[2 * i1], y1 = V[2 * i1 + 1];
    float x2 = V[2 * i2], y2 = V[2 * i2 + 1];
    // C = 2! * area = |cross|
    float cr = (x1 - x0) * (y2 - y0) - (x2 - x0) * (y1 - y0);
    CD[e] = fabsf(cr) * D[e];
    Px[4 * e + 0] = x0;  Py[4 * e + 0] = y0;
    Px[4 * e + 1] = x1;  Py[4 * e + 1] = y1;
    Px[4 * e + 2] = x2;  Py[4 * e + 2] = y2;
    Px[4 * e + 3] = 0.f; Py[4 * e + 3] = 0.f;   // pad row -> sig = 0, unused
}

// ------------------------------------------------------------------
// Kernel 2: main accumulation. 8 waves/block, 16 freq bins per wave.
// sig tile (16 ev-rows x 16 bins) produced by v_wmma_f32_16x16x4_f32.
// ------------------------------------------------------------------
__global__ __launch_bounds__(256) void ddsl_main(
    const float* __restrict__ gPx, const float* __restrict__ gPy,
    const float* __restrict__ gCD, float* __restrict__ out)
{
    __shared__ float sPx[4 * N_ELEM];
    __shared__ float sPy[4 * N_ELEM];
    __shared__ float sCD[N_ELEM];

    int tid = threadIdx.x;
    for (int i = tid; i < 4 * N_ELEM; i += 256) { sPx[i] = gPx[i]; sPy[i] = gPy[i]; }
    sCD[tid] = gCD[tid];
    __syncthreads();

    const int lane   = tid & 31;
    const int waveId = tid >> 5;
    const int half   = lane >> 4;     // 0: K-half 0 / elems {0,1};  1: K-half 1 / elems {2,3}
    const int n      = lane & 15;     // freq column within tile

    const int bin = (blockIdx.x * 8 + waveId) * 16 + n;
    const int u   = bin / RES1;
    const int v   = bin - u * RES1;
    const float fu = (u < 128) ? (float)u : (float)(u - 256);   // fftfreq * res
    const float wx = TWO_PI_F * fu;
    const float wy = TWO_PI_F * (float)v;                       // rfftfreq * res

    // B (4x16): rows K=0,1 = (wx, wy) on lanes 0-15; rows K=2,3 zero on lanes 16-31
    const float hmask = half ? 0.f : 1.f;
    v2f b;
    b.x = hmask * wx;
    b.y = hmask * wy;

    float sRe = 0.f, sIm = 0.f;

    for (int t = 0; t < N_ELEM / 4; ++t) {
        // A (16x4): lanes 0-15 hold K=0,1 = (px,py) of row M=lane;
        // lanes 16-31 hold the zero K=2,3 half. Unconditional LDS load
        // (address is in range for all lanes) + multiply-mask keeps EXEC
        // uniform through the WMMA.
        int ar = t * 16 + n;
        v2f a;
        a.x = hmask * sPx[ar];
        a.y = hmask * sPy[ar];

        v8f c = {};
        v8f d = __builtin_amdgcn_wmma_f32_16x16x4_f32(
            false, a, false, b, (short)0, c, false, false);
        // D layout: lane n (n<16) VGPR k -> sig[M=k, bin n]; lane 16+n -> sig[M=8+k, bin n]
        // => this lane owns elements eA = 4t+2*half (d[0..2]) and eB = eA+1 (d[4..6])

        #pragma unroll
        for (int q = 0; q < 2; ++q) {
            float s0 = d[4 * q + 0];
            float s1 = d[4 * q + 1];
            float s2 = d[4 * q + 2];
            float d01 = s0 - s1, d02 = s0 - s2, d12 = s1 - s2;
            float den0 = d01 * d02;          // (s0-s1)(s0-s2)
            float den1 = d12 * (-d01);       // (s1-s2)(s1-s0)
            float den2 = d02 * d12;          // (s2-s0)(s2-s1)
            den0 = (den0 == 0.f) ? 1.f : den0;
            den1 = (den1 == 0.f) ? 1.f : den1;
            den2 = (den2 == 0.f) ? 1.f : den2;
            // one v_rcp_f32 per denominator, shared by cos and sin terms
            float r0 = __builtin_amdgcn_rcpf(den0);
            float r1 = __builtin_amdgcn_rcpf(den1);
            float r2 = __builtin_amdgcn_rcpf(den2);
            float sn0, cs0, sn1, cs1, sn2, cs2;
            __sincosf(s0, &sn0, &cs0);
            __sincosf(s1, &sn1, &cs1);
            __sincosf(s2, &sn2, &cs2);
            float re = fmaf(cs0, r0, fmaf(cs1, r1, cs2 * r2));
            float im = fmaf(sn0, r0, fmaf(sn1, r1, sn2 * r2));
            float cd = sCD[t * 4 + 2 * half + q];
            sRe = fmaf(cd, re, sRe);
            sIm = fmaf(cd, im, sIm);
        }
    }

    // merge lane halves: elements {0,1 mod 4} (lanes 0-15) + {2,3 mod 4} (lanes 16-31)
    sRe += __shfl_xor(sRe, 16, 32);
    sIm += __shfl_xor(sIm, 16, 32);

    if (half == 0 && bin < NBINS) {
        // F = img(SumCD*(cos,-sin)/den, deg=2) * RES0^2 = (-65536*sRe, +65536*sIm)
        out[2 * bin + 0] = -65536.f * sRe;
        out[2 * bin + 1] =  65536.f * sIm;
    }
}

// ------------------------------------------------------------------
// Kernel 3: overwrite DC bin with (-1/2! * sum(CD)) * (-1) * RES0^2
// ------------------------------------------------------------------
__global__ __launch_bounds__(256) void ddsl_dc(const float* __restrict__ CD,
                                               float* __restrict__ out)
{
    __shared__ float red[256];
    int t = threadIdx.x;
    red[t] = CD[t];
    __syncthreads();
    for (int s = 128; s > 0; s >>= 1) {
        if (t < s) red[t] += red[t + s];
        __syncthreads();
    }
    if (t == 0) {
        float val = 32768.f * red[0];   // (1/2)*sum(CD)*65536
        out[0] = val;
        out[1] = val;
    }
}

// ------------------------------------------------------------------
extern "C" void kernel_launch(void* const* d_in, const int* in_sizes, int n_in,
                              void* d_out, int out_size, void* d_ws, size_t ws_size,
                              hipStream_t stream)
{
    const float* V = (const float*)d_in[0];
    const int*   E = (const int*)d_in[1];
    const float* D = (const float*)d_in[2];
    float* out = (float*)d_out;

    float* Px = (float*)d_ws;            // 1024 floats
    float* Py = Px + 4 * N_ELEM;         // 1024 floats
    float* CD = Py + 4 * N_ELEM;         // 256 floats  (total 9216 B of ws)

    ddsl_precompute<<<1, 256, 0, stream>>>(V, E, D, Px, Py, CD);

    int nblocks = (NBINS + 127) / 128;   // 258, exact cover of 33024 bins
    ddsl_main<<<nblocks, 256, 0, stream>>>(Px, Py, CD, out);

    ddsl_dc<<<1, 256, 0, stream>>>(CD, out);
}